// FinalGNNModel_35871566856412
// MI455X (gfx1250) — compile-verified
//
#include <hip/hip_runtime.h>
#include <math.h>

#define NNODE 1024
#define NEDGE 32768
#define HDIM  256
#define KDIM  512   // 2*H
#define NHEAD 4
#define NLAY  3
#define FDIM  517   // 2H + D
#define DDIM  5

typedef float v2f __attribute__((ext_vector_type(2)));
typedef float v8f __attribute__((ext_vector_type(8)));

// ---------------------------------------------------------------------------
// helpers
// ---------------------------------------------------------------------------
__device__ __forceinline__ float block_reduce_sum(float v, float* red, int t) {
  red[t] = v; __syncthreads();
  if (t < 64) red[t] += red[t + 64] + red[t + 128] + red[t + 192];
  __syncthreads();
  float r = 0.f;
  if (t < 32) {
    r = red[t] + red[t + 32];
    for (int off = 16; off > 0; off >>= 1) r += __shfl_down(r, off, 32);
  }
  __syncthreads();
  return r;   // valid in t==0
}

// Async global->LDS copy of 16 bytes (CDNA5 ASYNCcnt path, no VGPR staging).
// lds_off: wave-relative LDS byte offset (low 32 bits of generic pointer).
__device__ __forceinline__ void async_copy_b128(unsigned lds_off, const float* gp) {
  asm volatile("global_load_async_to_lds_b128 %0, %1, off"
               :: "v"(lds_off), "v"(gp) : "memory");
}
__device__ __forceinline__ void async_wait0() {
  asm volatile("s_wait_asynccnt 0" ::: "memory");
}

__global__ __launch_bounds__(256) void zero_kernel(float* __restrict__ p, int n) {
  int i = blockIdx.x * 256 + threadIdx.x;
  if (i < n) p[i] = 0.f;
}

// ---------------------------------------------------------------------------
// Edge GEMM (psi) with fused bias+ReLU+segment_sum scatter.
// Tile: 16 edges x 256 outputs per block (8 waves, each 2 WMMA col-tiles).
// A (16x4 f32) streamed from LDS-staged gathered features, B (4x16) as
// float2 loads from row-major psi_w (B[k,o]=w[o,k]; (k,k+1) pair == v0/v1).
// Feature rows staged with global_load_async_to_lds_b128.
// ---------------------------------------------------------------------------
__global__ __launch_bounds__(256) void edge_psi_kernel(
    const float* __restrict__ cur, const int* __restrict__ eidx,
    const float* __restrict__ w, const float* __restrict__ bias,
    float* __restrict__ aggr)
{
  __shared__ float feat[16][KDIM + 4];
  __shared__ int   sdst[16];
  const int tile = blockIdx.x;
  const int t = threadIdx.x;

  { // stage concat(cur[dst], cur[src]) rows for 16 edges into LDS (async)
    int e_local = t >> 4;
    int part    = t & 15;            // 16 threads/edge, 32 floats each
    int e   = tile * 16 + e_local;
    int src = eidx[e];               // edge_index[0] = src
    int dst = eidx[NEDGE + e];       // edge_index[1] = dst
    if (part == 0) sdst[e_local] = dst;
    const float* base = (part < 8) ? (cur + (size_t)dst * HDIM)
                                   : (cur + (size_t)src * HDIM);
    int off = (part & 7) * 32;
    const float* gp = base + off;
    float* lp = &feat[e_local][(part < 8 ? 0 : HDIM) + off];
    unsigned lds_off = (unsigned)(unsigned long long)lp;
#pragma unroll
    for (int j = 0; j < 8; ++j)
      async_copy_b128(lds_off + 16u * j, gp + 4 * j);
  }
  async_wait0();
  __syncthreads();

  const int wave = t >> 5, lane = t & 31;
  const int mN   = lane & 15;
  const int koff = (lane < 16) ? 0 : 2;     // A/B lane k-offset (K=4 layout)
  const int col0 = wave * 32;
  const float* wr0 = w + (size_t)(col0 + mN) * KDIM + koff;
  const float* wr1 = wr0 + (size_t)16 * KDIM;
  const float* ar  = &feat[mN][koff];

  v8f acc0 = {}; v8f acc1 = {};
#pragma unroll 4
  for (int kb = 0; kb < KDIM; kb += 4) {
    v2f Av = *(const v2f*)(ar  + kb);
    v2f B0 = *(const v2f*)(wr0 + kb);
    v2f B1 = *(const v2f*)(wr1 + kb);
    acc0 = __builtin_amdgcn_wmma_f32_16x16x4_f32(false, Av, false, B0,
                                                 (short)0, acc0, false, false);
    acc1 = __builtin_amdgcn_wmma_f32_16x16x4_f32(false, Av, false, B1,
                                                 (short)0, acc1, false, false);
  }

  const float b0 = bias[col0 + mN];
  const float b1 = bias[col0 + 16 + mN];
  const int erow = (lane < 16) ? 0 : 8;     // C layout: M = j + 8*(lane>=16)
#pragma unroll
  for (int j = 0; j < 8; ++j) {
    int d = sdst[erow + j];
    float v0 = acc0[j] + b0; v0 = v0 > 0.f ? v0 : 0.f;
    float v1 = acc1[j] + b1; v1 = v1 > 0.f ? v1 : 0.f;
    atomicAdd(aggr + (size_t)d * HDIM + col0 + mN,      v0);
    atomicAdd(aggr + (size_t)d * HDIM + col0 + 16 + mN, v1);
  }
}

// ---------------------------------------------------------------------------
// Node GEMM (phi): cur_next = relu(concat(cur, aggr) @ phi_w.T + b)
// ---------------------------------------------------------------------------
__global__ __launch_bounds__(256) void node_phi_kernel(
    const float* __restrict__ cur, const float* __restrict__ aggr,
    const float* __restrict__ w, const float* __restrict__ bias,
    float* __restrict__ outb)
{
  __shared__ float feat[16][KDIM + 4];
  const int tile = blockIdx.x;
  const int t = threadIdx.x;
  {
    int n_local = t >> 4;
    int part    = t & 15;
    int n = tile * 16 + n_local;
    const float* base = (part < 8) ? (cur + (size_t)n * HDIM)
                                   : (aggr + (size_t)n * HDIM);
    int off = (part & 7) * 32;
    const float* gp = base + off;
    float* lp = &feat[n_local][(part < 8 ? 0 : HDIM) + off];
    unsigned lds_off = (unsigned)(unsigned long long)lp;
#pragma unroll
    for (int j = 0; j < 8; ++j)
      async_copy_b128(lds_off + 16u * j, gp + 4 * j);
  }
  async_wait0();
  __syncthreads();

  const int wave = t >> 5, lane = t & 31;
  const int mN   = lane & 15;
  const int koff = (lane < 16) ? 0 : 2;
  const int col0 = wave * 32;
  const float* wr0 = w + (size_t)(col0 + mN) * KDIM + koff;
  const float* wr1 = wr0 + (size_t)16 * KDIM;
  const float* ar  = &feat[mN][koff];

  v8f acc0 = {}; v8f acc1 = {};
#pragma unroll 4
  for (int kb = 0; kb < KDIM; kb += 4) {
    v2f Av = *(const v2f*)(ar  + kb);
    v2f B0 = *(const v2f*)(wr0 + kb);
    v2f B1 = *(const v2f*)(wr1 + kb);
    acc0 = __builtin_amdgcn_wmma_f32_16x16x4_f32(false, Av, false, B0,
                                                 (short)0, acc0, false, false);
    acc1 = __builtin_amdgcn_wmma_f32_16x16x4_f32(false, Av, false, B1,
                                                 (short)0, acc1, false, false);
  }

  const float b0 = bias[col0 + mN];
  const float b1 = bias[col0 + 16 + mN];
  const int nrow = tile * 16 + ((lane < 16) ? 0 : 8);
#pragma unroll
  for (int j = 0; j < 8; ++j) {
    float v0 = acc0[j] + b0; v0 = v0 > 0.f ? v0 : 0.f;
    float v1 = acc1[j] + b1; v1 = v1 > 0.f ? v1 : 0.f;
    outb[(size_t)(nrow + j) * HDIM + col0 + mN]      = v0;
    outb[(size_t)(nrow + j) * HDIM + col0 + 16 + mN] = v1;
  }
}

// ---------------------------------------------------------------------------
// a[k,u,l] = wa[k].h[l][u], b[k,u,l] = wb[k].h[l][u]   (one block per node)
// ---------------------------------------------------------------------------
__global__ __launch_bounds__(256) void ab_kernel(
    const float* __restrict__ h0, const float* __restrict__ h1,
    const float* __restrict__ h2, const float* __restrict__ attn,
    float* __restrict__ a, float* __restrict__ b)
{
  __shared__ float red[256];
  const int u = blockIdx.x, t = threadIdx.x;
  const float hv0 = h0[(size_t)u * HDIM + t];
  const float hv1 = h1[(size_t)u * HDIM + t];
  const float hv2 = h2[(size_t)u * HDIM + t];
  for (int k = 0; k < NHEAD; ++k) {
    const float wa = attn[k * FDIM + t];
    const float wb = attn[k * FDIM + HDIM + t];
    float r;
    r = block_reduce_sum(hv0 * wa, red, t); if (t == 0) a[(k*NNODE+u)*3+0] = r;
    r = block_reduce_sum(hv1 * wa, red, t); if (t == 0) a[(k*NNODE+u)*3+1] = r;
    r = block_reduce_sum(hv2 * wa, red, t); if (t == 0) a[(k*NNODE+u)*3+2] = r;
    r = block_reduce_sum(hv0 * wb, red, t); if (t == 0) b[(k*NNODE+u)*3+0] = r;
    r = block_reduce_sum(hv1 * wb, red, t); if (t == 0) b[(k*NNODE+u)*3+1] = r;
    r = block_reduce_sum(hv2 * wb, red, t); if (t == 0) b[(k*NNODE+u)*3+2] = r;
  }
}

// per-head shifts: shifts[k]=max a, shifts[4+k]=max b, shifts[8+k]=max wc
__global__ __launch_bounds__(256) void shift_kernel(
    const float* __restrict__ a, const float* __restrict__ b,
    const float* __restrict__ attn, float* __restrict__ shifts)
{
  __shared__ float red[256];
  const int t = threadIdx.x;
  for (int k = 0; k < NHEAD; ++k) {
    float m = -3.0e38f;
    for (int i = t; i < NNODE * 3; i += 256) m = fmaxf(m, a[k*NNODE*3 + i]);
    red[t] = m; __syncthreads();
    for (int s = 128; s > 0; s >>= 1) { if (t < s) red[t] = fmaxf(red[t], red[t+s]); __syncthreads(); }
    if (t == 0) shifts[k] = red[0];
    __syncthreads();
    m = -3.0e38f;
    for (int i = t; i < NNODE * 3; i += 256) m = fmaxf(m, b[k*NNODE*3 + i]);
    red[t] = m; __syncthreads();
    for (int s = 128; s > 0; s >>= 1) { if (t < s) red[t] = fmaxf(red[t], red[t+s]); __syncthreads(); }
    if (t == 0) {
      shifts[4 + k] = red[0];
      float mc = -3.0e38f;
      for (int d = 0; d < DDIM; ++d) mc = fmaxf(mc, attn[k*FDIM + 2*HDIM + d]);
      shifts[8 + k] = mc;
    }
    __syncthreads();
  }
}

// E[k,u,l] = exp(ab - shift_k); Sum[k,u] = sum_l E
__global__ __launch_bounds__(256) void exp_kernel(
    const float* __restrict__ ab, const float* __restrict__ shifts, int shoff,
    float* __restrict__ E, float* __restrict__ Sum)
{
  const int idx = blockIdx.x * 256 + threadIdx.x;     // k*N + u
  if (idx >= NHEAD * NNODE) return;
  const int k = idx / NNODE;
  const float sh = shifts[shoff + k];
  float s = 0.f;
#pragma unroll
  for (int l = 0; l < 3; ++l) {
    float e = __expf(ab[idx*3 + l] - sh);
    E[idx*3 + l] = e;
    s += e;
  }
  Sum[idx] = s;
}

// Row pass over dist: Trow[k,d] = sum_{v!=u, dm1==d} B[k,v]
// -> Cu[k,u] = sum_d ec*Trow ; S[k,d] += A[k,u]*Trow (atomics, 20 per block)
__global__ __launch_bounds__(256) void row_kernel(
    const int* __restrict__ dist, const float* __restrict__ Bsum,
    const float* __restrict__ Asum, const float* __restrict__ attn,
    const float* __restrict__ shifts, float* __restrict__ Cu, float* __restrict__ S)
{
  __shared__ float red[256];
  const int u = blockIdx.x, t = threadIdx.x;
  float tr[NHEAD][DDIM] = {};
  for (int v = t; v < NNODE; v += 256) {
    if (v == u) continue;
    const int dm = dist[(size_t)u * NNODE + v] - 1;   // 0..4
    float bv[NHEAD];
#pragma unroll
    for (int k = 0; k < NHEAD; ++k) bv[k] = Bsum[k*NNODE + v];
#pragma unroll
    for (int d = 0; d < DDIM; ++d) {
      const float msk = (dm == d) ? 1.f : 0.f;
#pragma unroll
      for (int k = 0; k < NHEAD; ++k) tr[k][d] += msk * bv[k];
    }
  }
  float sums[NHEAD * DDIM];
#pragma unroll
  for (int i = 0; i < NHEAD * DDIM; ++i) {
    red[t] = tr[i / DDIM][i % DDIM]; __syncthreads();
    if (t < 64) red[t] += red[t+64] + red[t+128] + red[t+192];
    __syncthreads();
    if (t == 0) { float s2 = 0.f; for (int j = 0; j < 64; ++j) s2 += red[j]; sums[i] = s2; }
    __syncthreads();
  }
  if (t == 0) {
    for (int k = 0; k < NHEAD; ++k) {
      const float av = Asum[k*NNODE + u];
      float cu = 0.f;
      for (int d = 0; d < DDIM; ++d) {
        const float trv = sums[k*DDIM + d];
        const float ec  = __expf(attn[k*FDIM + 2*HDIM + d] - shifts[8 + k]);
        cu += ec * trv;
        atomicAdd(&S[k*DDIM + d], av * trv);
      }
      Cu[k*NNODE + u] = cu;
    }
  }
}

// Column pass: Cv[k,v] = sum_d ec * sum_{u!=v, dm1==d} A[k,u]
__global__ __launch_bounds__(256) void col_kernel(
    const int* __restrict__ dist, const float* __restrict__ Asum,
    const float* __restrict__ attn, const float* __restrict__ shifts,
    float* __restrict__ Cv)
{
  __shared__ float red[256];
  const int v = blockIdx.x, t = threadIdx.x;
  float tr[NHEAD][DDIM] = {};
  for (int u = t; u < NNODE; u += 256) {
    if (u == v) continue;
    const int dm = dist[(size_t)u * NNODE + v] - 1;
    float av[NHEAD];
#pragma unroll
    for (int k = 0; k < NHEAD; ++k) av[k] = Asum[k*NNODE + u];
#pragma unroll
    for (int d = 0; d < DDIM; ++d) {
      const float msk = (dm == d) ? 1.f : 0.f;
#pragma unroll
      for (int k = 0; k < NHEAD; ++k) tr[k][d] += msk * av[k];
    }
  }
  float sums[NHEAD * DDIM];
#pragma unroll
  for (int i = 0; i < NHEAD * DDIM; ++i) {
    red[t] = tr[i / DDIM][i % DDIM]; __syncthreads();
    if (t < 64) red[t] += red[t+64] + red[t+128] + red[t+192];
    __syncthreads();
    if (t == 0) { float s2 = 0.f; for (int j = 0; j < 64; ++j) s2 += red[j]; sums[i] = s2; }
    __syncthreads();
  }
  if (t == 0) {
    for (int k = 0; k < NHEAD; ++k) {
      float cv = 0.f;
      for (int d = 0; d < DDIM; ++d)
        cv += __expf(attn[k*FDIM + 2*HDIM + d] - shifts[8 + k]) * sums[k*DDIM + d];
      Cv[k*NNODE + v] = cv;
    }
  }
}

// part[k,h] = sum_u C[k,u] * sum_l E[k,u,l] * h_l[u,h]   (denom applied later)
__global__ __launch_bounds__(256) void part_kernel(
    const float* __restrict__ E, const float* __restrict__ C,
    const float* __restrict__ h0, const float* __restrict__ h1,
    const float* __restrict__ h2, float* __restrict__ part)
{
  const int k = blockIdx.x, t = threadIdx.x;     // t = h column
  float acc = 0.f;
  for (int u = 0; u < NNODE; ++u) {
    const float c  = C[k*NNODE + u];
    const float w0 = E[(k*NNODE + u)*3 + 0] * c;
    const float w1 = E[(k*NNODE + u)*3 + 1] * c;
    const float w2 = E[(k*NNODE + u)*3 + 2] * c;
    acc += w0 * h0[(size_t)u*HDIM + t]
         + w1 * h1[(size_t)u*HDIM + t]
         + w2 * h2[(size_t)u*HDIM + t];
  }
  part[k*HDIM + t] = acc;
}

// finale: denom, part3, feat = [p1,p2,p3]/denom, out = sigmoid(sum attn*feat*fw + fb)
__global__ __launch_bounds__(256) void final_kernel(
    const float* __restrict__ part1, const float* __restrict__ part2,
    const float* __restrict__ S, const float* __restrict__ attn,
    const float* __restrict__ shifts, const float* __restrict__ fw,
    const float* __restrict__ fb, float* __restrict__ out)
{
  __shared__ float red[256];
  __shared__ float sp3[NHEAD * DDIM];
  __shared__ float sden[NHEAD];
  const int t = threadIdx.x;
  if (t < NHEAD * DDIM) {
    const int k = t / DDIM, d = t % DDIM;
    sp3[t] = __expf(attn[k*FDIM + 2*HDIM + d] - shifts[8 + k]) * S[k*DDIM + d];
  }
  __syncthreads();
  if (t < NHEAD) {
    float dsum = 0.f;
    for (int d = 0; d < DDIM; ++d) dsum += sp3[t*DDIM + d];
    sden[t] = dsum;
  }
  __syncthreads();
  float acc = 0.f;
  for (int idx = t; idx < NHEAD * FDIM; idx += 256) {
    const int k = idx / FDIM, j = idx % FDIM;
    float feat;
    if (j < HDIM)          feat = part1[k*HDIM + j] / sden[k];
    else if (j < 2*HDIM)   feat = part2[k*HDIM + (j - HDIM)] / sden[k];
    else                   feat = sp3[k*DDIM + (j - 2*HDIM)] / sden[k];
    acc += attn[idx] * feat * fw[idx];
  }
  const float tot = block_reduce_sum(acc, red, t);
  if (t == 0) out[0] = 1.f / (1.f + __expf(-(tot + fb[0])));
}

// ---------------------------------------------------------------------------
extern "C" void kernel_launch(void* const* d_in, const int* in_sizes, int n_in,
                              void* d_out, int out_size, void* d_ws, size_t ws_size,
                              hipStream_t stream) {
  (void)in_sizes; (void)n_in; (void)out_size; (void)ws_size;
  const float* x     = (const float*)d_in[0];
  const int*   eidx  = (const int*)d_in[1];
  const int*   dist  = (const int*)d_in[3];
  const float* psi_w = (const float*)d_in[4];
  const float* psi_b = (const float*)d_in[5];
  const float* phi_w = (const float*)d_in[6];
  const float* phi_b = (const float*)d_in[7];
  const float* attn  = (const float*)d_in[8];
  const float* fw    = (const float*)d_in[9];
  const float* fb    = (const float*)d_in[10];
  float* out = (float*)d_out;

  float* ws     = (float*)d_ws;
  float* h1     = ws;                          // N*H
  float* h2     = h1   + NNODE * HDIM;
  float* aggr   = h2   + NNODE * HDIM;
  float* aA     = aggr + NNODE * HDIM;         // K*N*3
  float* bB     = aA   + NHEAD * NNODE * 3;
  float* EA     = bB   + NHEAD * NNODE * 3;
  float* EB     = EA   + NHEAD * NNODE * 3;
  float* Asum   = EB   + NHEAD * NNODE * 3;    // K*N
  float* Bsum   = Asum + NHEAD * NNODE;
  float* Cu     = Bsum + NHEAD * NNODE;
  float* Cv     = Cu   + NHEAD * NNODE;
  float* Sbuf   = Cv   + NHEAD * NNODE;        // 20 (+pad)
  float* shifts = Sbuf + 32;                   // 12 (+pad)
  float* part1  = shifts + 16;                 // K*H
  float* part2  = part1 + NHEAD * HDIM;

  const int WLAY = HDIM * KDIM;                // 131072 per layer

  // ---- layer 0 (cur = x) ----
  zero_kernel<<<NNODE * HDIM / 256, 256, 0, stream>>>(aggr, NNODE * HDIM);
  edge_psi_kernel<<<NEDGE / 16, 256, 0, stream>>>(x, eidx, psi_w, psi_b, aggr);
  node_phi_kernel<<<NNODE / 16, 256, 0, stream>>>(x, aggr, phi_w, phi_b, h1);
  // ---- layer 1 (cur = h1) ----  (layer 2 output is unused by the reference)
  zero_kernel<<<NNODE * HDIM / 256, 256, 0, stream>>>(aggr, NNODE * HDIM);
  edge_psi_kernel<<<NEDGE / 16, 256, 0, stream>>>(h1, eidx, psi_w + WLAY, psi_b + HDIM, aggr);
  node_phi_kernel<<<NNODE / 16, 256, 0, stream>>>(h1, aggr, phi_w + WLAY, phi_b + HDIM, h2);

  // ---- attention (separable softmax) ----
  ab_kernel<<<NNODE, 256, 0, stream>>>(x, h1, h2, attn, aA, bB);
  shift_kernel<<<1, 256, 0, stream>>>(aA, bB, attn, shifts);
  exp_kernel<<<NHEAD * NNODE / 256, 256, 0, stream>>>(aA, shifts, 0, EA, Asum);
  exp_kernel<<<NHEAD * NNODE / 256, 256, 0, stream>>>(bB, shifts, 4, EB, Bsum);
  zero_kernel<<<1, 256, 0, stream>>>(Sbuf, NHEAD * DDIM);
  row_kernel<<<NNODE, 256, 0, stream>>>(dist, Bsum, Asum, attn, shifts, Cu, Sbuf);
  col_kernel<<<NNODE, 256, 0, stream>>>(dist, Asum, attn, shifts, Cv);
  part_kernel<<<NHEAD, 256, 0, stream>>>(EA, Cu, x, h1, h2, part1);
  part_kernel<<<NHEAD, 256, 0, stream>>>(EB, Cv, x, h1, h2, part2);
  final_kernel<<<1, 256, 0, stream>>>(part1, part2, Sbuf, attn, shifts, fw, fb, out);
}